// RiemannianBatchNorm_36850819399681
// MI455X (gfx1250) — compile-verified
//
#include <hip/hip_runtime.h>
#include <cmath>

#define EPSF 1e-15f
#define N_ROWS 65536
#define DIM 64
#define TPB 256
#define NBLK 256
#define WAVES_PER_BLK (TPB / 32)
#define TOTAL_WAVES (NBLK * WAVES_PER_BLK)
#define FRECHET_ITERS 10

typedef float v2f __attribute__((ext_vector_type(2)));
typedef float v8f __attribute__((ext_vector_type(8)));

// ---------- wave32 helpers ----------
__device__ __forceinline__ float wredsum(float v) {
#pragma unroll
  for (int off = 16; off > 0; off >>= 1) v += __shfl_xor(v, off, 32);
  return v;  // uniform across the 32 lanes
}

__device__ __forceinline__ float artanhf_(float u) {
  u = fminf(fmaxf(u, -1.0f + 1e-7f), 1.0f - 1e-7f);
  return 0.5f * (log1pf(u) - log1pf(-u));
}

// ---------- WMMA block combine ----------
// lds holds 8 per-wave partial 64-vectors (lds[w*64 + d]).  Executed by wave 0
// with full EXEC.  Column sums of the 8x64 matrix via V_WMMA_F32_16X16X4_F32:
// D = ones(16x4) x B(4x16) + C.  B layout (f32, 2 VGPRs):
//   VGPR0: lanes 0-15 -> B[0][n], lanes 16-31 -> B[2][n]
//   VGPR1: lanes 0-15 -> B[1][n], lanes 16-31 -> B[3][n]
// D layout: C VGPR0, lane n(<16) = D[0][n] = colsum(n).
__device__ __forceinline__ void block_combine_wmma(const float* lds, float* out,
                                                   int lane) {
  const v2f ones = {1.0f, 1.0f};
  const int col = (lane & 15);
  const int hi = (lane >> 4) & 1;  // 0: lanes 0-15, 1: lanes 16-31
#pragma unroll
  for (int t = 0; t < 4; ++t) {
    const int c16 = t * 16 + col;
    v2f b0, b1;
    b0.x = lds[(0 + 2 * hi) * DIM + c16];  // wave partials 0..3 as K rows
    b0.y = lds[(1 + 2 * hi) * DIM + c16];
    b1.x = lds[(4 + 2 * hi) * DIM + c16];  // wave partials 4..7 as K rows
    b1.y = lds[(5 + 2 * hi) * DIM + c16];
    v8f c = {};
    c = __builtin_amdgcn_wmma_f32_16x16x4_f32(false, ones, false, b0, (short)0,
                                              c, false, false);
    c = __builtin_amdgcn_wmma_f32_16x16x4_f32(false, ones, false, b1, (short)0,
                                              c, false, false);
    if (lane < 16) out[t * 16 + lane] = c[0];
  }
}

// ---------- tiny kernels (1 wave) ----------
__global__ void k_init_onm(const float* __restrict__ mean,
                           float* __restrict__ ws_onm) {
  const int lane = threadIdx.x;  // 32 threads
  float vx = mean[2 * lane], vy = mean[2 * lane + 1];
  float n2 = wredsum(vx * vx + vy * vy);
  float nv = fmaxf(sqrtf(n2), EPSF);
  float f = tanhf(nv) / nv;
  ws_onm[2 * lane] = f * vx;
  ws_onm[2 * lane + 1] = f * vy;
}

__global__ void k_mu_init(const float* __restrict__ part,
                          float* __restrict__ ws_mu) {
  const int lane = threadIdx.x;
  float sx = 0.f, sy = 0.f;
  for (int b = 0; b < NBLK; ++b) {
    sx += part[b * DIM + 2 * lane];
    sy += part[b * DIM + 2 * lane + 1];
  }
  const float inv = 1.0f / (float)N_ROWS;
  float vx = sx * inv, vy = sy * inv;
  float n2 = wredsum(vx * vx + vy * vy);
  float nv = fmaxf(sqrtf(n2), EPSF);
  float f = tanhf(nv) / nv;
  ws_mu[2 * lane] = f * vx;
  ws_mu[2 * lane + 1] = f * vy;
}

__global__ void k_mu_update(const float* __restrict__ part,
                            float* __restrict__ ws_mu) {
  const int lane = threadIdx.x;
  float sx = 0.f, sy = 0.f;
  for (int b = 0; b < NBLK; ++b) {
    sx += part[b * DIM + 2 * lane];
    sy += part[b * DIM + 2 * lane + 1];
  }
  const float inv = 1.0f / (float)N_ROWS;
  float vx = sx * inv, vy = sy * inv;
  float mx = ws_mu[2 * lane], my = ws_mu[2 * lane + 1];
  float m2 = wredsum(mx * mx + my * my);
  float v2 = wredsum(vx * vx + vy * vy);
  float mv = wredsum(mx * vx + my * vy);
  float nv = fmaxf(sqrtf(v2), EPSF);
  float lam = 2.0f / fmaxf(1.0f - m2, EPSF);
  float t = tanhf(0.5f * lam * nv) / nv;  // w = t*v
  float w2 = t * t * v2;
  float mw = t * mv;
  float den = 1.0f + 2.0f * mw + m2 * w2;
  float invd = 1.0f / fmaxf(den, EPSF);
  float ca = (1.0f + 2.0f * mw + w2) * invd;
  float cb = (1.0f - m2) * invd * t;
  ws_mu[2 * lane] = ca * mx + cb * vx;
  ws_mu[2 * lane + 1] = ca * my + cb * vy;
}

__global__ void k_finalize(const float* __restrict__ sparts,
                           const float* __restrict__ var1,
                           const float* __restrict__ ws_mu,
                           const float* __restrict__ ws_onm,
                           float* __restrict__ ws_scal) {
  const int lane = threadIdx.x;
  float mx = ws_mu[2 * lane], my = ws_mu[2 * lane + 1];
  float ox = ws_onm[2 * lane], oy = ws_onm[2 * lane + 1];
  float m2 = wredsum(mx * mx + my * my);
  float o2 = wredsum(ox * ox + oy * oy);
  float s = 0.f;
  for (int b = 0; b < NBLK; ++b) s += sparts[b];
  float input_var = s / (float)N_ROWS;
  float scale = sqrtf(var1[0] / (input_var + 1e-6f));
  float lam_m = 2.0f / fmaxf(1.0f - m2, EPSF);
  float lam_o = 2.0f / fmaxf(1.0f - o2, EPSF);
  if (lane == 0) ws_scal[0] = (lam_m / lam_o) * scale;
}

// ---------- heavy reduction kernels ----------
__global__ void k_partial_log0(const float* __restrict__ x,
                               float* __restrict__ part) {
  __shared__ float lds[WAVES_PER_BLK * DIM];
  const int lane = threadIdx.x & 31;
  const int wid = threadIdx.x >> 5;
  const int g = blockIdx.x * WAVES_PER_BLK + wid;
  float ax = 0.f, ay = 0.f;
  for (int r = g; r < N_ROWS; r += TOTAL_WAVES) {
    const float* row = x + (size_t)r * DIM + 2 * lane;
    __builtin_prefetch(row + (size_t)TOTAL_WAVES * DIM, 0, 0);
    float vx = row[0], vy = row[1];
    float n2 = wredsum(vx * vx + vy * vy);
    float nx = fmaxf(sqrtf(n2), EPSF);
    float f = artanhf_(nx) / nx;
    ax += f * vx;
    ay += f * vy;
  }
  lds[wid * DIM + 2 * lane] = ax;
  lds[wid * DIM + 2 * lane + 1] = ay;
  __syncthreads();
  if (threadIdx.x < 32)
    block_combine_wmma(lds, part + (size_t)blockIdx.x * DIM, lane);
}

__global__ void k_partial_logmap(const float* __restrict__ x,
                                 const float* __restrict__ ws_mu,
                                 float* __restrict__ part) {
  __shared__ float lds[WAVES_PER_BLK * DIM];
  const int lane = threadIdx.x & 31;
  const int wid = threadIdx.x >> 5;
  const int g = blockIdx.x * WAVES_PER_BLK + wid;
  const float mx = ws_mu[2 * lane], my = ws_mu[2 * lane + 1];
  const float m2 = wredsum(mx * mx + my * my);
  const float two_over_lam = fmaxf(1.0f - m2, EPSF);  // 2/lam(mu)
  float ax = 0.f, ay = 0.f;
  for (int r = g; r < N_ROWS; r += TOTAL_WAVES) {
    const float* row = x + (size_t)r * DIM + 2 * lane;
    __builtin_prefetch(row + (size_t)TOTAL_WAVES * DIM, 0, 0);
    float vx = row[0], vy = row[1];
    float y2 = wredsum(vx * vx + vy * vy);
    float ab = -wredsum(mx * vx + my * vy);  // dot(-mu, x)
    float den = 1.0f + 2.0f * ab + m2 * y2;
    float invd = 1.0f / fmaxf(den, EPSF);
    float ca = (1.0f + 2.0f * ab + y2) * invd;  // coef of (-mu)
    float cb = (1.0f - m2) * invd;              // coef of x
    float d2 = fmaxf(ca * ca * m2 + 2.0f * ca * cb * ab + cb * cb * y2, 0.f);
    float nd = fmaxf(sqrtf(d2), EPSF);
    float gc = two_over_lam * artanhf_(nd) / nd;
    ax += gc * (cb * vx - ca * mx);
    ay += gc * (cb * vy - ca * my);
  }
  lds[wid * DIM + 2 * lane] = ax;
  lds[wid * DIM + 2 * lane + 1] = ay;
  __syncthreads();
  if (threadIdx.x < 32)
    block_combine_wmma(lds, part + (size_t)blockIdx.x * DIM, lane);
}

__global__ void k_partial_var(const float* __restrict__ x,
                              const float* __restrict__ ws_mu,
                              float* __restrict__ sparts) {
  __shared__ float lds[WAVES_PER_BLK];
  const int lane = threadIdx.x & 31;
  const int wid = threadIdx.x >> 5;
  const int g = blockIdx.x * WAVES_PER_BLK + wid;
  const float mx = ws_mu[2 * lane], my = ws_mu[2 * lane + 1];
  const float m2 = wredsum(mx * mx + my * my);
  float acc = 0.f;
  for (int r = g; r < N_ROWS; r += TOTAL_WAVES) {
    const float* row = x + (size_t)r * DIM + 2 * lane;
    float vx = row[0], vy = row[1];
    float y2 = wredsum(vx * vx + vy * vy);
    float ab = -wredsum(mx * vx + my * vy);
    float den = 1.0f + 2.0f * ab + m2 * y2;
    float invd = 1.0f / fmaxf(den, EPSF);
    float ca = (1.0f + 2.0f * ab + y2) * invd;
    float cb = (1.0f - m2) * invd;
    float d2 = fmaxf(ca * ca * m2 + 2.0f * ca * cb * ab + cb * cb * y2, 0.f);
    float dist = 2.0f * artanhf_(sqrtf(d2));
    acc += dist * dist;
  }
  if (lane == 0) lds[wid] = acc;
  __syncthreads();
  if (threadIdx.x == 0) {
    float s = 0.f;
#pragma unroll
    for (int w = 0; w < WAVES_PER_BLK; ++w) s += lds[w];
    sparts[blockIdx.x] = s;
  }
}

// ---------- final transform ----------
// Every intermediate vector is a linear combination of {mu, x, onm}; track
// 3 coefficients and use the Gram matrix -> only 3 shuffle reductions / row.
struct V3 {
  float m, x, o;
};
struct Gram {
  float m2, y2, o2, mx, om, ox;
};

__device__ __forceinline__ float dot3(const V3& a, const V3& b, const Gram& G) {
  return a.m * b.m * G.m2 + a.x * b.x * G.y2 + a.o * b.o * G.o2 +
         (a.m * b.x + a.x * b.m) * G.mx + (a.m * b.o + a.o * b.m) * G.om +
         (a.x * b.o + a.o * b.x) * G.ox;
}

__device__ __forceinline__ V3 mob3(const V3& a, const V3& b, const Gram& G) {
  float a2 = dot3(a, a, G), b2 = dot3(b, b, G), ab = dot3(a, b, G);
  float den = 1.0f + 2.0f * ab + a2 * b2;
  float inv = 1.0f / fmaxf(den, EPSF);
  float ca = (1.0f + 2.0f * ab + b2) * inv;
  float cb = (1.0f - a2) * inv;
  V3 r;
  r.m = ca * a.m + cb * b.m;
  r.x = ca * a.x + cb * b.x;
  r.o = ca * a.o + cb * b.o;
  return r;
}

__global__ void k_final(const float* __restrict__ x,
                        const float* __restrict__ ws_mu,
                        const float* __restrict__ ws_onm,
                        const float* __restrict__ ws_scal,
                        float* __restrict__ out) {
  const int lane = threadIdx.x & 31;
  const int wid = threadIdx.x >> 5;
  const int g = blockIdx.x * WAVES_PER_BLK + wid;
  const float mx = ws_mu[2 * lane], my = ws_mu[2 * lane + 1];
  const float ox = ws_onm[2 * lane], oy = ws_onm[2 * lane + 1];
  const float coef = ws_scal[0];
  const float m2 = wredsum(mx * mx + my * my);
  const float o2 = wredsum(ox * ox + oy * oy);
  const float om = wredsum(mx * ox + my * oy);
  const float two_over_lam_m = fmaxf(1.0f - m2, EPSF);
  const float lam_o = 2.0f / fmaxf(1.0f - o2, EPSF);
  const V3 NM = {-1.f, 0.f, 0.f}, X = {0.f, 1.f, 0.f}, O = {0.f, 0.f, 1.f};
  for (int r = g; r < N_ROWS; r += TOTAL_WAVES) {
    const float* row = x + (size_t)r * DIM + 2 * lane;
    __builtin_prefetch(row + (size_t)TOTAL_WAVES * DIM, 0, 0);
    float vx = row[0], vy = row[1];
    Gram G;
    G.m2 = m2; G.o2 = o2; G.om = om;
    G.y2 = wredsum(vx * vx + vy * vy);
    G.mx = wredsum(mx * vx + my * vy);
    G.ox = wredsum(ox * vx + oy * vy);
    // l = logmap(mu, x)
    V3 d = mob3(NM, X, G);
    float d2 = fmaxf(dot3(d, d, G), 0.f);
    float nd = fmaxf(sqrtf(d2), EPSF);
    float gc = two_over_lam_m * artanhf_(nd) / nd;
    V3 l = {gc * d.m, gc * d.x, gc * d.o};
    // gyr(onm, -mu, l) = mob(-mob(onm,-mu), mob(onm, mob(-mu, l)))
    V3 t1 = mob3(NM, l, G);
    V3 t2 = mob3(O, t1, G);
    V3 p = mob3(O, NM, G);
    V3 q = {-p.m, -p.x, -p.o};
    V3 t3 = mob3(q, t2, G);
    // logm = (lam(mu)/lam(onm)) * sqrt(var/(input_var+1e-6)) * gyr = coef*t3
    V3 w = {coef * t3.m, coef * t3.x, coef * t3.o};
    float w2 = fmaxf(dot3(w, w, G), 0.f);
    float nw = fmaxf(sqrtf(w2), EPSF);
    float te = tanhf(0.5f * lam_o * nw) / nw;
    V3 z = {te * w.m, te * w.x, te * w.o};
    V3 res = mob3(O, z, G);  // expmap(onm, logm)
    float* orow = out + (size_t)r * DIM + 2 * lane;
    orow[0] = res.m * mx + res.x * vx + res.o * ox;
    orow[1] = res.m * my + res.x * vy + res.o * oy;
  }
}

extern "C" void kernel_launch(void* const* d_in, const int* in_sizes, int n_in,
                              void* d_out, int out_size, void* d_ws,
                              size_t ws_size, hipStream_t stream) {
  const float* x = (const float*)d_in[0];     // [65536, 64]
  const float* mean = (const float*)d_in[1];  // [64]
  const float* var = (const float*)d_in[2];   // [1]
  float* out = (float*)d_out;                 // [65536, 64]

  float* W = (float*)d_ws;
  float* ws_mu = W;                          // 64
  float* ws_onm = W + 64;                    // 64
  float* ws_scal = W + 128;                  // 8
  float* ws_part = W + 136;                  // NBLK*64
  float* ws_sp = W + 136 + NBLK * DIM;       // NBLK

  k_init_onm<<<1, 32, 0, stream>>>(mean, ws_onm);
  k_partial_log0<<<NBLK, TPB, 0, stream>>>(x, ws_part);
  k_mu_init<<<1, 32, 0, stream>>>(ws_part, ws_mu);
  for (int i = 0; i < FRECHET_ITERS; ++i) {
    k_partial_logmap<<<NBLK, TPB, 0, stream>>>(x, ws_mu, ws_part);
    k_mu_update<<<1, 32, 0, stream>>>(ws_part, ws_mu);
  }
  k_partial_var<<<NBLK, TPB, 0, stream>>>(x, ws_mu, ws_sp);
  k_finalize<<<1, 32, 0, stream>>>(ws_sp, var, ws_mu, ws_onm, ws_scal);
  k_final<<<NBLK, TPB, 0, stream>>>(x, ws_mu, ws_onm, ws_scal, out);
}